// AbstentionCTC_80118319940191
// MI455X (gfx1250) — compile-verified
//
#include <hip/hip_runtime.h>
#include <math.h>

// Abstention-CTC loss (forward DP + analytic adjoint sweep) for gfx1250.
// log_probs (T,B,V) f32, targets (B,L) i32, input_lengths (B) i32,
// target_lengths (B) i32.  V fixed at 128 by problem spec; T derived.
//
// Workspace: alpha[B][T][Sp] f32 (Sp = S rounded to 8), loss[B], partial[B].

#define NEG_INF (-__builtin_inff())

// ---------------- CDNA5 async global->LDS path ----------------
#if defined(__has_builtin)
#  if __has_builtin(__builtin_amdgcn_global_load_async_to_lds_b32) && \
      __has_builtin(__builtin_amdgcn_s_wait_asynccnt)
#    define USE_ASYNC_LDS 1
#  endif
#endif
#ifndef USE_ASYNC_LDS
#  define USE_ASYNC_LDS 0
#endif

#if USE_ASYNC_LDS
typedef int __attribute__((address_space(1))) as1_i32;   // global scalar
typedef int __attribute__((address_space(3))) as3_i32;   // LDS scalar
typedef int v4i __attribute__((__vector_size__(4 * sizeof(int))));
typedef v4i __attribute__((address_space(1))) as1_v4i;   // global 4xi32
typedef v4i __attribute__((address_space(3))) as3_v4i;   // LDS 4xi32
#  if __has_builtin(__builtin_amdgcn_global_load_async_to_lds_b128)
#    define HAVE_ASYNC_B128 1
#  else
#    define HAVE_ASYNC_B128 0
#  endif
#endif

// Copy n floats global->LDS. Requires blockDim.x >= ceil(n/4) when the b128
// path is used (true here: max n = 404, blockDim = 416), and 16B alignment of
// both pointers (guaranteed by padded strides).
__device__ __forceinline__ void lds_copy_async(float* dst_lds, const float* src_glb,
                                               int n, int tid) {
#if USE_ASYNC_LDS
#  if HAVE_ASYNC_B128
  const int n4 = n >> 2;
  if (tid < n4)
    __builtin_amdgcn_global_load_async_to_lds_b128(
        (as1_v4i*)(src_glb + 4 * tid), (as3_v4i*)(dst_lds + 4 * tid), 0, 0);
  const int rem = n & 3;
  if (tid < rem)
    __builtin_amdgcn_global_load_async_to_lds_b32(
        (as1_i32*)(src_glb + 4 * n4 + tid), (as3_i32*)(dst_lds + 4 * n4 + tid), 0, 0);
#  else
  if (tid < n)
    __builtin_amdgcn_global_load_async_to_lds_b32(
        (as1_i32*)(src_glb + tid), (as3_i32*)(dst_lds + tid), 0, 0);
#  endif
#else
  if (tid < n) dst_lds[tid] = src_glb[tid];
#endif
}

__device__ __forceinline__ void lds_copy_wait() {
#if USE_ASYNC_LDS
  __builtin_amdgcn_s_wait_asynccnt(0);   // drain ASYNCcnt before the barrier
#endif
}

// ---------------- math helpers ----------------
__device__ __forceinline__ float logaddexp_f(float a, float b) {
  float m = fmaxf(a, b);
  if (m == NEG_INF) return NEG_INF;
  float d = fminf(a, b) - m;              // may be -inf -> exp -> 0
  return m + __logf(1.0f + __expf(d));
}

__device__ __forceinline__ float lse3_f(float a, float b, float c) {
  float m = fmaxf(fmaxf(a, b), c);
  if (m == NEG_INF) return NEG_INF;
  float s = __expf(a - m) + __expf(b - m) + __expf(c - m);
  return m + __logf(s);
}

// ---------------- forward DP ----------------
__global__ void actc_forward(const float* __restrict__ lp, const int* __restrict__ targets,
                             const int* __restrict__ ilen_p, const int* __restrict__ tlen_p,
                             float* __restrict__ alpha_g, float* __restrict__ loss_ws,
                             int T, int B, int V, int L, int S, int Sp) {
  const int b = blockIdx.x;
  const int tid = threadIdx.x;
  const int nthr = blockDim.x;
  extern __shared__ float smem[];
  float* lbuf = smem;                 // 2*V  (double-buffered lp row, 16B-aligned)
  float* abuf = lbuf + 2 * V;         // 2*S  (double-buffered alpha)
  int* tp = (int*)(abuf + 2 * S);     // S
  int* sk = tp + S;                   // S

  for (int s = tid; s < S; s += nthr)
    tp[s] = (s & 1) ? targets[(size_t)b * L + (s >> 1)] : 0 /*BLANK*/;
  __syncthreads();
  for (int s = tid; s < S; s += nthr) {
    int v = tp[s];
    int p2 = (s >= 2) ? tp[s - 2] : -1;
    sk[s] = (v != 0) && (v != p2);
  }
  lds_copy_async(lbuf, lp + (size_t)b * V, V, tid);   // row 0
  lds_copy_wait();
  __syncthreads();

  const int il = ilen_p[b];
  { // alpha row 0
    float lpA = lbuf[1];
    if (tid < S) {
      float a = NEG_INF;
      if (tid == 0)      a = logaddexp_f(lbuf[tp[0]], lpA);  // stay[0,0]
      else if (tid == 1) a = lbuf[tp[1]];                    // em[0,1]
      abuf[tid] = a;
      alpha_g[((size_t)b * T) * Sp + tid] = a;
    }
  }
  if (T > 1) lds_copy_async(lbuf + V, lp + ((size_t)B + b) * V, V, tid);

  for (int t = 1; t < T; ++t) {
    lds_copy_wait();
    __syncthreads();                                       // lp[t] + alpha[t-1] visible
    if (t + 1 < T)
      lds_copy_async(lbuf + ((t + 1) & 1) * V, lp + ((size_t)(t + 1) * B + b) * V, V, tid);
    const float* lrow = lbuf + (t & 1) * V;
    const float* aprev = abuf + ((t - 1) & 1) * S;
    float* acur = abuf + (t & 1) * S;
    if (tid < S) {
      const int s = tid;
      float an;
      if (t < il) {
        float em = lrow[tp[s]];
        float stay = logaddexp_f(em, lrow[1]);             // ABST = 1
        float av = stay + aprev[s];
        float bv = (s >= 1) ? em + aprev[s - 1] : NEG_INF;
        float cv = (sk[s] && s >= 2) ? em + aprev[s - 2] : NEG_INF;
        an = lse3_f(av, bv, cv);
      } else {
        an = aprev[s];
      }
      acur[s] = an;
      alpha_g[((size_t)b * T + t) * Sp + s] = an;
    }
  }
  __syncthreads();
  if (tid == 0) {
    const float* afin = abuf + ((T - 1) & 1) * S;
    int idx = 2 * tlen_p[b];
    int idx2 = (idx > 0) ? idx - 1 : 0;
    loss_ws[b] = logaddexp_f(afin[idx], afin[idx2]);
  }
}

// ---------------- adjoint sweep: directly accumulates sum(grads * lp) ----------------
__global__ void actc_backward(const float* __restrict__ lp, const int* __restrict__ targets,
                              const int* __restrict__ ilen_p, const int* __restrict__ tlen_p,
                              const float* __restrict__ alpha_g, const float* __restrict__ loss_ws,
                              float* __restrict__ partial_ws,
                              int T, int B, int V, int L, int S, int Sp, int Sa) {
  const int b = blockIdx.x;
  const int tid = threadIdx.x;
  const int nthr = blockDim.x;
  extern __shared__ float smem[];
  float* lbuf = smem;                  // 2*V
  float* arow = lbuf + 2 * V;          // 3*Sa (alpha rows, index = t mod 3; Sa keeps 16B align)
  float* gbuf = arow + 3 * Sa;         // 2*Sa (adjoint double buffer)
  float* wa = gbuf + 2 * Sa;           // Sa
  float* wb = wa + Sa;                 // Sa
  float* wc = wb + Sa;                 // Sa
  float* red = wc + Sa;                // 32
  int* tp = (int*)(red + 32);          // S
  int* sk = tp + S;                    // S

  for (int s = tid; s < S; s += nthr)
    tp[s] = (s & 1) ? targets[(size_t)b * L + (s >> 1)] : 0;
  __syncthreads();
  for (int s = tid; s < S; s += nthr) {
    int v = tp[s];
    int p2 = (s >= 2) ? tp[s - 2] : -1;
    sk[s] = (v != 0) && (v != p2);
  }

  const int il = ilen_p[b];
  const float loss = loss_ws[b];
  const float* Abase = alpha_g + (size_t)b * T * Sp;

  // prologue: alpha row T-1, init adjoints g = softmax over terminal states
  lds_copy_async(arow + ((T - 1) % 3) * Sa, Abase + (size_t)(T - 1) * Sp, S, tid);
  lds_copy_wait();
  __syncthreads();
  int par = 0;
  if (tid < S) {
    float av = arow[((T - 1) % 3) * Sa + tid];
    float g = 0.0f;
    int idx = 2 * tlen_p[b];
    if ((tid == idx || tid == idx - 1) && loss > NEG_INF && av > NEG_INF)
      g = __expf(av - loss);
    gbuf[par * Sa + tid] = g;
  }
  lds_copy_async(lbuf + ((T - 1) & 1) * V, lp + ((size_t)(T - 1) * B + b) * V, V, tid);
  if (T >= 2)
    lds_copy_async(arow + ((T - 2) % 3) * Sa, Abase + (size_t)(T - 2) * Sp, S, tid);

  float acc = 0.0f;

  for (int t = T - 1; t >= 1; --t) {
    lds_copy_wait();
    __syncthreads();
    // prefetch for iteration t-1
    lds_copy_async(lbuf + ((t - 1) & 1) * V, lp + ((size_t)(t - 1) * B + b) * V, V, tid);
    if (t - 2 >= 0)
      lds_copy_async(arow + ((t - 2) % 3) * Sa, Abase + (size_t)(t - 2) * Sp, S, tid);

    const float* lrow = lbuf + (t & 1) * V;
    const float* ats = arow + (t % 3) * Sa;         // alpha[t]
    const float* apv = arow + ((t - 1) % 3) * Sa;   // alpha[t-1]
    const float* gcur = gbuf + par * Sa;

    if (tid < S) {
      const int s = tid;
      float g = gcur[s];
      float ga = 0.0f, gb = 0.0f, gc = 0.0f;
      if (t >= il) {
        ga = g;                                    // alpha passed through unchanged
      } else if (g > 0.0f) {
        float at = ats[s];
        if (at > NEG_INF) {
          float em = lrow[tp[s]];
          float lA = lrow[1];
          float stay = logaddexp_f(em, lA);
          float av = stay + apv[s];
          float bv = (s >= 1) ? em + apv[s - 1] : NEG_INF;
          float cv = (sk[s] && s >= 2) ? em + apv[s - 2] : NEG_INF;
          ga = g * __expf(av - at);
          gb = (bv > NEG_INF) ? g * __expf(bv - at) : 0.0f;
          gc = (cv > NEG_INF) ? g * __expf(cv - at) : 0.0f;
          // grads.lp contributions: d_em*em + d_abst*lpA
          float pe = __expf(em - stay);
          float pa = __expf(lA - stay);
          acc += (gb + gc + ga * pe) * em + ga * pa * lA;
        }
      }
      wa[s] = ga; wb[s] = gb; wc[s] = gc;
    }
    __syncthreads();
    if (tid < S) {   // gather adjoints for alpha[t-1]
      const int s = tid;
      float gn = wa[s];
      if (s + 1 < S) gn += wb[s + 1];
      if (s + 2 < S) gn += wc[s + 2];
      gbuf[(par ^ 1) * Sa + s] = gn;
    }
    par ^= 1;
  }

  // epilogue: t = 0 initialization terms (lp row 0 is in lbuf[0])
  lds_copy_wait();
  __syncthreads();
  {
    const float* l0 = lbuf;
    const float* g0 = gbuf + par * Sa;
    if (tid == 0) {
      float em = l0[tp[0]];
      float lA = l0[1];
      float stay = logaddexp_f(em, lA);
      acc += g0[0] * (__expf(em - stay) * em + __expf(lA - stay) * lA);
    } else if (tid == 1 && S > 1) {
      acc += g0[1] * l0[tp[1]];
    }
  }

  // block reduction (wave32 shuffles, then LDS)
  float v = acc;
  for (int off = 16; off > 0; off >>= 1) v += __shfl_down(v, off, 32);
  if ((tid & 31) == 0) red[tid >> 5] = v;
  __syncthreads();
  if (tid == 0) {
    float s = 0.0f;
    int nw = (nthr + 31) >> 5;
    for (int w = 0; w < nw; ++w) s += red[w];
    partial_ws[b] = s;
  }
}

__global__ void actc_finalize(const float* __restrict__ partial_ws, float* __restrict__ out, int B) {
  if (threadIdx.x == 0) {
    float s = 0.0f;
    for (int i = 0; i < B; ++i) s += partial_ws[i];
    out[0] = s / (float)B / -4.0f;
  }
}

extern "C" void kernel_launch(void* const* d_in, const int* in_sizes, int n_in,
                              void* d_out, int out_size, void* d_ws, size_t ws_size,
                              hipStream_t stream) {
  (void)n_in; (void)out_size; (void)ws_size;
  const float* lp = (const float*)d_in[0];
  const int* targets = (const int*)d_in[1];
  const int* ilen = (const int*)d_in[2];
  const int* tlen = (const int*)d_in[3];

  const int B = in_sizes[2];
  const int L = in_sizes[1] / B;
  const int V = 128;                       // fixed by problem spec
  const int T = in_sizes[0] / (B * V);
  const int S = 2 * L + 1;
  const int Sp = (S + 7) & ~7;             // global row stride (32B aligned)
  const int Sa = (S + 3) & ~3;             // LDS row stride (16B aligned)

  float* alpha_ws = (float*)d_ws;
  size_t alpha_elems = (size_t)B * T * Sp;
  float* loss_ws = alpha_ws + alpha_elems;
  float* partial_ws = loss_ws + B;

  int nthr = ((S + 31) / 32) * 32;         // 416 = 13 wave32
  if (nthr < ((V + 31) / 32) * 32) nthr = ((V + 31) / 32) * 32;

  size_t smemF = (size_t)(2 * V + 2 * S) * sizeof(float) + (size_t)(2 * S) * sizeof(int);
  size_t smemB = (size_t)(2 * V + 3 * Sa + 2 * Sa + 3 * Sa + 32) * sizeof(float) +
                 (size_t)(2 * S) * sizeof(int);

  actc_forward<<<B, nthr, smemF, stream>>>(lp, targets, ilen, tlen, alpha_ws, loss_ws,
                                           T, B, V, L, S, Sp);
  actc_backward<<<B, nthr, smemB, stream>>>(lp, targets, ilen, tlen, alpha_ws, loss_ws,
                                            partial_ws, T, B, V, L, S, Sp, Sa);
  actc_finalize<<<1, 32, 0, stream>>>(partial_ws, (float*)d_out, B);
}